// MultiHeadMemory_84318797955729
// MI455X (gfx1250) — compile-verified
//
#include <hip/hip_runtime.h>
#include <stdint.h>

// MI455X / gfx1250: wave32, V_WMMA_F32_16X16X4_F32 (exact fp32 GEMM path),
// Tensor Data Mover (TDM) for global->LDS staging with hardware padding.
// Memory-bound (AI ~16 flop/byte): stream K,V,Q exactly once from HBM.

typedef __attribute__((ext_vector_type(2))) float        v2f;
typedef __attribute__((ext_vector_type(8))) float        v8f;
typedef __attribute__((ext_vector_type(4))) unsigned int u32x4;
typedef __attribute__((ext_vector_type(4))) int          i32x4;
typedef __attribute__((ext_vector_type(8))) int          i32x8;

#define WPAD  65         // LDS row pitch in floats; produced by TDM pad feature
#define EPSF  1e-6f
#define SPLIT 8          // S-dimension split for phase 1
#define QROWS 256        // Q rows per block in phase 2

#define WMMA_F32(A, B, C) \
    __builtin_amdgcn_wmma_f32_16x16x4_f32(false, (A), false, (B), (short)0, (C), false, false)

__device__ __forceinline__ uint32_t lds_addr32(const void* p)
{
    // generic pointer to __shared__: low 32 bits are the LDS byte offset
    return (uint32_t)(uintptr_t)p;
}

// ---------------------------------------------------------------------------
// TDM: DMA `rows` x 64 floats (row-major, global row stride 64 floats) into
// LDS, inserting 1 pad dword after every 64 dwords -> LDS pitch = WPAD = 65.
// Issue from one wave only; completion via s_wait_tensorcnt.
// D# encoding per CDNA5 ISA 8.3/8.4.
// ---------------------------------------------------------------------------
__device__ __forceinline__ void tdm_load_rows64(uint32_t lds_bytes, const float* g, int rows)
{
    const uint64_t ga = (uint64_t)(uintptr_t)g;

    u32x4 g0;
    g0.x = 1u;                                                  // count=1, user D#
    g0.y = lds_bytes;                                           // lds_addr
    g0.z = (uint32_t)ga;                                        // global_addr[31:0]
    g0.w = (uint32_t)((ga >> 32) & 0x1FFFFFFu) | (2u << 30);    // addr[56:32] | type=2

    i32x8 g1;
    g1[0] = (2 << 16)                 // data_size = 4 bytes
          | (1 << 20)                 // pad_enable
          | (5 << 22)                 // pad_interval: 2^(5+1) = 64 dwords
          | (0 << 25);                // pad_amount: 1 dword
    g1[1] = (64 & 0xFFFF) << 16;                      // tensor_dim0 = 64
    g1[2] = (rows & 0xFFFF) << 16;                    // tensor_dim1 = rows
    g1[3] = ((rows >> 16) & 0xFFFF) | (64 << 16);     // tile_dim0 = 64
    g1[4] = (rows & 0xFFFF);                          // tile_dim1 = rows, tile_dim2 = 0
    g1[5] = 64;                                       // tensor_dim0_stride = 64
    g1[6] = 0;
    g1[7] = 0;

    i32x4 g2 = {0, 0, 0, 0};
    i32x4 g3 = {0, 0, 0, 0};

#if defined(__clang_major__) && (__clang_major__ >= 23)
    i32x8 g4 = {0, 0, 0, 0, 0, 0, 0, 0};
    __builtin_amdgcn_tensor_load_to_lds(g0, g1, g2, g3, g4, 0);
#else
    __builtin_amdgcn_tensor_load_to_lds(g0, g1, g2, g3, 0);
#endif
}

// ---------------------------------------------------------------------------
// Kernel 1: partial M = K^T V and z = colsum(K) over an S/SPLIT slice.
// TDM double-buffers 64x64 K/V chunks (DMA of chunk c+1 overlaps compute of
// chunk c). 16 waves = 4 k-groups x (2x2 waves); each wave owns a 32x32 block
// of M with 4 accumulators (4 WMMAs per 4 fragment-pair loads).
// ---------------------------------------------------------------------------
__global__ __launch_bounds__(512)
void mhm_build_kernel(const float* __restrict__ K,
                      const float* __restrict__ V,
                      float* __restrict__ Mpart,
                      float* __restrict__ Zpart,
                      int S)
{
    __shared__ float Kb2[2][64 * WPAD];
    __shared__ float Vb2[2][64 * WPAD];   // Vb2[0] reused as reduce buffer at end
    __shared__ float zp[512];

    const int bh   = blockIdx.x;
    const int sp   = blockIdx.y;
    const int BH   = gridDim.x;
    const int t    = threadIdx.x;
    const int lane = t & 31;
    const int wave = t >> 5;
    const int grp  = wave >> 2;                // k-step group 0..3
    const int pos  = wave & 3;
    const int rowpos = (pos >> 1) * 32;        // 32x32 block position in M
    const int colpos = (pos & 1) * 32;

    // 32-bit A 16x4 fragment layout: lanes 0..15 -> M=lane, K=kk+{0,1};
    //                                lanes 16..31 -> M=lane-16, K=kk+{2,3}.
    const int mrow = lane & 15;
    const int koff = (lane < 16) ? 0 : 2;
    const int aCol0 = rowpos + mrow,  aCol1 = rowpos + 16 + mrow;
    const int bCol0 = colpos + mrow,  bCol1 = colpos + 16 + mrow;

    const int Sslice = S / SPLIT;
    const int nc     = Sslice / 64;            // chunks in this slice
    const float* __restrict__ Kbase = K + (size_t)bh * S * 64 + (size_t)sp * Sslice * 64;
    const float* __restrict__ Vbase = V + (size_t)bh * S * 64 + (size_t)sp * Sslice * 64;

    // prologue: wave 0 DMAs chunk 0 into buffer 0
    if (t < 32) {
        tdm_load_rows64(lds_addr32(&Kb2[0][0]), Kbase, 64);
        tdm_load_rows64(lds_addr32(&Vb2[0][0]), Vbase, 64);
    }

    v8f a00 = {}, a01 = {}, a10 = {}, a11 = {};
    float zacc = 0.0f;
    const int zd = t & 63;
    const int zr = (t >> 6) * 8;

    for (int c = 0; c < nc; ++c) {
        if (t < 32) __builtin_amdgcn_s_wait_tensorcnt(0);   // chunk c landed
        __syncthreads();
        if ((t < 32) && (c + 1 < nc)) {                      // prefetch chunk c+1
            tdm_load_rows64(lds_addr32(&Kb2[(c + 1) & 1][0]),
                            Kbase + (size_t)(c + 1) * 64 * 64, 64);
            tdm_load_rows64(lds_addr32(&Vb2[(c + 1) & 1][0]),
                            Vbase + (size_t)(c + 1) * 64 * 64, 64);
        }
        const float* __restrict__ Kt = Kb2[c & 1];
        const float* __restrict__ Vt = Vb2[c & 1];

        // z partial over this chunk
#pragma unroll
        for (int q = 0; q < 8; ++q)
            zacc += Kt[(zr + q) * WPAD + zd];

        // this group's 4 k-steps: 4 frag pairs feed 4 WMMAs each
        const int kbase = grp * 16;
#pragma unroll
        for (int ks = 0; ks < 16; ks += 4) {
            const int kk = kbase + ks;
            v2f A0, A1, B0, B1;
            A0.x = Kt[(kk + koff) * WPAD + aCol0]; A0.y = Kt[(kk + koff + 1) * WPAD + aCol0];
            A1.x = Kt[(kk + koff) * WPAD + aCol1]; A1.y = Kt[(kk + koff + 1) * WPAD + aCol1];
            B0.x = Vt[(kk + koff) * WPAD + bCol0]; B0.y = Vt[(kk + koff + 1) * WPAD + bCol0];
            B1.x = Vt[(kk + koff) * WPAD + bCol1]; B1.y = Vt[(kk + koff + 1) * WPAD + bCol1];
            a00 = WMMA_F32(A0, B0, a00);
            a01 = WMMA_F32(A0, B1, a01);
            a10 = WMMA_F32(A1, B0, a10);
            a11 = WMMA_F32(A1, B1, a11);
        }
        __syncthreads();   // compute(c) done before buf[c&1] is overwritten at c+2
    }

    // deterministic z reduction (8 fixed-order partials per column)
    zp[t] = zacc;
    __syncthreads();
    if (t < 64) {
        float s = 0.0f;
#pragma unroll
        for (int j = 0; j < 8; ++j) s += zp[t + 64 * j];
        Zpart[((size_t)sp * BH + bh) * 64 + t] = s;
    }

    // reduce the 4 k-groups' partial Ms into group 0 via LDS (fixed order)
    float* red = &Vb2[0][0];   // 4096-float scratch, free after last barrier
#pragma unroll
    for (int rnd = 1; rnd < 4; ++rnd) {
        if (grp == rnd) {
#pragma unroll
            for (int r = 0; r < 8; ++r) {
                const int rr0 = rowpos +      ((lane < 16) ? r : 8 + r);
                const int rr1 = rowpos + 16 + ((lane < 16) ? r : 8 + r);
                red[rr0 * 64 + colpos + mrow]      = a00[r];
                red[rr0 * 64 + colpos + 16 + mrow] = a01[r];
                red[rr1 * 64 + colpos + mrow]      = a10[r];
                red[rr1 * 64 + colpos + 16 + mrow] = a11[r];
            }
        }
        __syncthreads();
        if (grp == 0) {
#pragma unroll
            for (int r = 0; r < 8; ++r) {
                const int rr0 = rowpos +      ((lane < 16) ? r : 8 + r);
                const int rr1 = rowpos + 16 + ((lane < 16) ? r : 8 + r);
                a00[r] += red[rr0 * 64 + colpos + mrow];
                a01[r] += red[rr0 * 64 + colpos + 16 + mrow];
                a10[r] += red[rr1 * 64 + colpos + mrow];
                a11[r] += red[rr1 * 64 + colpos + 16 + mrow];
            }
        }
        __syncthreads();
    }

    // group 0 writes this slice's partial M
    if (grp == 0) {
        float* __restrict__ Mo = Mpart + ((size_t)sp * BH + bh) * 4096;
#pragma unroll
        for (int r = 0; r < 8; ++r) {
            const int rr0 = rowpos +      ((lane < 16) ? r : 8 + r);
            const int rr1 = rowpos + 16 + ((lane < 16) ? r : 8 + r);
            Mo[rr0 * 64 + colpos + mrow]      = a00[r];
            Mo[rr0 * 64 + colpos + 16 + mrow] = a01[r];
            Mo[rr1 * 64 + colpos + mrow]      = a10[r];
            Mo[rr1 * 64 + colpos + 16 + mrow] = a11[r];
        }
    }
}

// ---------------------------------------------------------------------------
// Kernel 1b: sum the SPLIT partial M/z slices (L2-resident, ~9 MB traffic).
// ---------------------------------------------------------------------------
__global__ __launch_bounds__(256)
void mhm_reduce_kernel(const float* __restrict__ Mpart,
                       const float* __restrict__ Zpart,
                       float* __restrict__ Mf,
                       float* __restrict__ Zf)
{
    const int bh = blockIdx.x;
    const int BH = gridDim.x;
    const int t  = threadIdx.x;

    for (int e = t; e < 4096; e += 256) {
        float s = 0.0f;
#pragma unroll
        for (int sp = 0; sp < SPLIT; ++sp)
            s += Mpart[((size_t)sp * BH + bh) * 4096 + e];
        Mf[(size_t)bh * 4096 + e] = s;
    }
    if (t < 64) {
        float s = 0.0f;
#pragma unroll
        for (int sp = 0; sp < SPLIT; ++sp)
            s += Zpart[((size_t)sp * BH + bh) * 64 + t];
        Zf[bh * 64 + t] = s;
    }
}

// ---------------------------------------------------------------------------
// Kernel 2: out = (Q.M) * rcp(Q.z + eps) for a 256-row Q chunk.
// TDM stages Q (256x64) and M (64x64); 16 waves each compute a 32x32 block.
// ---------------------------------------------------------------------------
__global__ __launch_bounds__(512)
void mhm_apply_kernel(const float* __restrict__ Q,
                      const float* __restrict__ Mf,
                      const float* __restrict__ Zf,
                      float* __restrict__ Out,
                      int S)
{
    __shared__ float Qt[QROWS * WPAD];
    __shared__ float Mt[64 * WPAD];
    __shared__ float zs[64];
    __shared__ float pden[512];
    __shared__ float den[QROWS];

    const int sc   = blockIdx.x * QROWS;
    const int bh   = blockIdx.y;
    const int t    = threadIdx.x;
    const int lane = t & 31;
    const int wave = t >> 5;
    const int rowpos = (wave >> 1) * 32;       // 8 row positions
    const int colpos = (wave & 1) * 32;        // 2 col positions

    const int mrow = lane & 15;
    const int koff = (lane < 16) ? 0 : 2;
    const int aRow0 = rowpos + mrow,  aRow1 = rowpos + 16 + mrow;
    const int bCol0 = colpos + mrow,  bCol1 = colpos + 16 + mrow;

    const float* __restrict__ Qb = Q  + (size_t)bh * S * 64 + (size_t)sc * 64;
    const float* __restrict__ Mb = Mf + (size_t)bh * 4096;

    // wave 0 DMAs Q chunk and M into LDS; barrier publishes to all waves
    if (t < 32) {
        tdm_load_rows64(lds_addr32(Qt), Qb, QROWS);
        tdm_load_rows64(lds_addr32(Mt), Mb, 64);
        __builtin_amdgcn_s_wait_tensorcnt(0);
    }
    if (t < 64) zs[t] = Zf[bh * 64 + t];
    __syncthreads();

    // den partial: 2 threads per row, 32 dims each (fixed order)
    {
        const int dr = t >> 1;
        const int dp = (t & 1) * 32;
        float p = 0.0f;
#pragma unroll
        for (int q = 0; q < 32; ++q)
            p += Qt[dr * WPAD + dp + q] * zs[dp + q];
        pden[t] = p;
    }

    // WMMA: 16 k-steps x 4 WMMAs (2x2 register block), ~1:1 ds:wmma
    v8f c00 = {}, c01 = {}, c10 = {}, c11 = {};
#pragma unroll
    for (int kk = 0; kk < 64; kk += 4) {
        v2f A0, A1, B0, B1;
        A0.x = Qt[aRow0 * WPAD + kk + koff];   A0.y = Qt[aRow0 * WPAD + kk + koff + 1];
        A1.x = Qt[aRow1 * WPAD + kk + koff];   A1.y = Qt[aRow1 * WPAD + kk + koff + 1];
        B0.x = Mt[(kk + koff) * WPAD + bCol0]; B0.y = Mt[(kk + koff + 1) * WPAD + bCol0];
        B1.x = Mt[(kk + koff) * WPAD + bCol1]; B1.y = Mt[(kk + koff + 1) * WPAD + bCol1];
        c00 = WMMA_F32(A0, B0, c00);
        c01 = WMMA_F32(A0, B1, c01);
        c10 = WMMA_F32(A1, B0, c10);
        c11 = WMMA_F32(A1, B1, c11);
    }
    __syncthreads();

    // deterministic den finish: one reciprocal per row
    if (t < QROWS)
        den[t] = 1.0f / (EPSF + pden[2 * t] + pden[2 * t + 1]);
    __syncthreads();

    // scale and store
    float* __restrict__ Ob = Out + (size_t)bh * S * 64 + (size_t)sc * 64;
#pragma unroll
    for (int r = 0; r < 8; ++r) {
        const int rr0 = rowpos +      ((lane < 16) ? r : 8 + r);
        const int rr1 = rowpos + 16 + ((lane < 16) ? r : 8 + r);
        Ob[(size_t)rr0 * 64 + colpos + mrow]      = c00[r] * den[rr0];
        Ob[(size_t)rr0 * 64 + colpos + 16 + mrow] = c01[r] * den[rr0];
        Ob[(size_t)rr1 * 64 + colpos + mrow]      = c10[r] * den[rr1];
        Ob[(size_t)rr1 * 64 + colpos + 16 + mrow] = c11[r] * den[rr1];
    }
}

// ---------------------------------------------------------------------------
extern "C" void kernel_launch(void* const* d_in, const int* in_sizes, int n_in,
                              void* d_out, int out_size, void* d_ws, size_t ws_size,
                              hipStream_t stream)
{
    (void)in_sizes; (void)n_in; (void)out_size; (void)ws_size;
    const float* keys    = (const float*)d_in[0];
    const float* values  = (const float*)d_in[1];
    const float* queries = (const float*)d_in[2];
    float* out = (float*)d_out;

    const int B = 4, H = 16, S = 4096, D = 64;
    const int BH = B * H;

    // workspace: partial M/z per split, then final M/z  (~9.6 MB total)
    float* Mpart = (float*)d_ws;                              // [SPLIT][BH][64*64]
    float* Zpart = Mpart + (size_t)SPLIT * BH * D * D;        // [SPLIT][BH][64]
    float* Mf    = Zpart + (size_t)SPLIT * BH * D;            // [BH][64*64]
    float* Zf    = Mf    + (size_t)BH * D * D;                // [BH][64]

    mhm_build_kernel <<<dim3(BH, SPLIT),     dim3(512), 0, stream>>>(keys, values, Mpart, Zpart, S);
    mhm_reduce_kernel<<<dim3(BH),            dim3(256), 0, stream>>>(Mpart, Zpart, Mf, Zf);
    mhm_apply_kernel <<<dim3(S / QROWS, BH), dim3(512), 0, stream>>>(queries, Mf, Zf, out, S);
}